// ADMMBlock_83227876262324
// MI455X (gfx1250) — compile-verified
//
#include <hip/hip_runtime.h>
#include <stdint.h>

// Problem constants (match reference)
#define T_LEN 12
#define NND   20000
#define KNN   10
#define KU    9
#define NH    4
#define NC    4
#define N_ADMM 3
#define N_CG   3

#define TNH      (T_LEN * NND * NH)        // 960,000 float4-groups
#define S_ELEMS  (TNH * NC)                // 3,840,000 floats per state tensor
#define SCAT_THR ((T_LEN - 1) * NND * NH)  // 880,000 scatter-target groups
#define E_EDGES  (NND * KNN)               // 200,000 graph edges

typedef __attribute__((ext_vector_type(2))) float v2f;
typedef __attribute__((ext_vector_type(8))) float v8f;

__device__ __forceinline__ float4 ld4(const float* p) { return *(const float4*)p; }
__device__ __forceinline__ void st4(float* p, float4 v) { *(float4*)p = v; }
__device__ __forceinline__ float4 zero4() { return make_float4(0.f, 0.f, 0.f, 0.f); }
__device__ __forceinline__ float4 fma4(float a, float4 x, float4 acc) {
  acc.x = fmaf(a, x.x, acc.x); acc.y = fmaf(a, x.y, acc.y);
  acc.z = fmaf(a, x.z, acc.z); acc.w = fmaf(a, x.w, acc.w);
  return acc;
}
__device__ __forceinline__ float4 add4(float4 a, float4 b) {
  return make_float4(a.x + b.x, a.y + b.y, a.z + b.z, a.w + b.w);
}
__device__ __forceinline__ float4 sub4(float4 a, float4 b) {
  return make_float4(a.x - b.x, a.y - b.y, a.z - b.z, a.w - b.w);
}
__device__ __forceinline__ float4 scale4(float s, float4 a) {
  return make_float4(s * a.x, s * a.y, s * a.z, s * a.w);
}

// ---------------------------------------------------------------- CSR build
__global__ void k_count(unsigned* __restrict__ deg, const int* __restrict__ nnf) {
  int e = blockIdx.x * blockDim.x + threadIdx.x;
  if (e >= E_EDGES) return;
  atomicAdd(&deg[nnf[e]], 1u);
}

// Single-workgroup exclusive scan over NND counts -> rowptr[NND+1]
__global__ void k_scan(unsigned* __restrict__ rowptr, const unsigned* __restrict__ deg) {
  __shared__ unsigned sh[1024];
  __shared__ unsigned carry;
  int tid = threadIdx.x;
  if (tid == 0) carry = 0;
  __syncthreads();
  for (int base = 0; base < NND; base += 1024) {
    unsigned v = (base + tid < NND) ? deg[base + tid] : 0u;
    sh[tid] = v;
    __syncthreads();
    for (int off = 1; off < 1024; off <<= 1) {
      unsigned t = (tid >= off) ? sh[tid - off] : 0u;
      __syncthreads();
      sh[tid] += t;
      __syncthreads();
    }
    unsigned excl = sh[tid] - v;
    if (base + tid < NND) rowptr[base + tid] = carry + excl;
    __syncthreads();
    if (tid == 1023) carry += sh[1023];
    __syncthreads();
  }
  if (tid == 0) rowptr[NND] = carry;
}

__global__ void k_fill(int* __restrict__ col, unsigned* __restrict__ cursor,
                       const int* __restrict__ nnf) {
  int e = blockIdx.x * blockDim.x + threadIdx.x;
  if (e >= E_EDGES) return;
  unsigned pos = atomicAdd(&cursor[nnf[e]], 1u);
  col[pos] = e;  // e = n*KNN + k
}

// ---------------------------------------------------------------- init
__global__ void k_init(float* __restrict__ x, float* __restrict__ gamma,
                       const float* __restrict__ y) {
  int id = blockIdx.x * blockDim.x + threadIdx.x;  // one (t,n,h) float4 per thread
  if (id >= TNH) return;
  int tn = id >> 2;
  float4 yv = ld4(y + (size_t)tn * NC);
  st4(x + (size_t)id * 4, yv);
  st4(gamma + (size_t)id * 4, make_float4(0.1f, 0.1f, 0.1f, 0.1f));
}

// ---------------------------------------------------------------- ldr (gather)
__global__ void k_ldr(float* __restrict__ out, const float* __restrict__ in,
                      const float* __restrict__ d_ew, const int* __restrict__ nn) {
  int id = blockIdx.x * blockDim.x + threadIdx.x;
  if (id >= TNH) return;
  int h = id & 3;
  int tn = id >> 2;
  int n = tn % NND;
  int t = tn / NND;
  float* op = out + (size_t)id * 4;
  if (t == 0) { st4(op, zero4()); return; }
  float4 acc = ld4(in + (size_t)id * 4);
  const int* nnr = nn + n * KNN;
  const float* dw = d_ew + ((size_t)((t - 1) * NND + n) * KNN) * NH + h;
  const float* prev = in + (size_t)(t - 1) * NND * 16;
#pragma unroll
  for (int k = 0; k < KNN; ++k) {
    int m = nnr[k];
    if (k + 1 < KNN)  // hide L2 latency of the next random gather
      __builtin_prefetch(prev + ((size_t)nnr[k + 1] * NH + h) * NC, 0, 3);
    float w = dw[k * NH];
    float4 g = ld4(prev + ((size_t)m * NH + h) * NC);
    acc = fma4(-w, g, acc);
  }
  st4(op, acc);
}

// ------------------------------------------- lhs non-scatter part:
//   out = hth(v) + mu_u*lu(v) + c2*ldr(v)[t]   (ldr result passed in `tmp`)
__global__ void k_lhs_ns(float* __restrict__ out, const float* __restrict__ v,
                         const float* __restrict__ tmp,
                         const float* __restrict__ u_ew, const int* __restrict__ nn,
                         const float* __restrict__ mu_u, const float* __restrict__ mu_d2,
                         const float* __restrict__ rho, const int* __restrict__ maskp,
                         int it) {
  int id = blockIdx.x * blockDim.x + threadIdx.x;
  if (id >= TNH) return;
  int h = id & 3;
  int tn = id >> 2;
  int n = tn % NND;
  int t = tn / NND;
  float muu = mu_u[it];
  float c2 = mu_d2[it] + 0.5f * rho[it];
  int mask = *maskp;
  float4 vc = ld4(v + (size_t)id * 4);
  float4 lu = vc;
  const int* nnr = nn + n * KNN + 1;  // nn[:,1:]
  const float* uw = u_ew + ((size_t)tn * KU) * NH + h;
  const float* vt = v + (size_t)t * NND * 16;
#pragma unroll
  for (int k = 0; k < KU; ++k) {
    int m = nnr[k];
    float w = uw[k * NH];
    float4 g = ld4(vt + ((size_t)m * NH + h) * NC);
    lu = fma4(-w, g, lu);
  }
  float4 res = (t < mask) ? vc : zero4();
  res = fma4(muu, lu.x == lu.x ? lu : lu, res);  // res += muu*lu
  float4 tz = ld4(tmp + (size_t)id * 4);         // tmp[0]==0 already
  res = fma4(c2, tz, res);
  st4(out + (size_t)id * 4, res);
}

// ------------------------------------------- lhs scatter part via reverse CSR:
//   out[t,m,h] -= c2 * sum_{e->m} d_ew[t,n_e,k_e,h] * tmp[t+1,n_e,h]
__global__ void k_lhs_scat(float* __restrict__ out, const float* __restrict__ tmp,
                           const float* __restrict__ d_ew,
                           const unsigned* __restrict__ rowptr, const int* __restrict__ col,
                           const float* __restrict__ mu_d2, const float* __restrict__ rho,
                           int it) {
  int id = blockIdx.x * blockDim.x + threadIdx.x;
  if (id >= SCAT_THR) return;
  int h = id & 3;
  int tm = id >> 2;
  int m = tm % NND;
  int t = tm / NND;  // 0..T-2
  float c2 = mu_d2[it] + 0.5f * rho[it];
  float4 acc = zero4();
  unsigned beg = rowptr[m], end = rowptr[m + 1];
  const float* dwb = d_ew + (size_t)t * NND * KNN * NH + h;
  const float* tp = tmp + (size_t)(t + 1) * NND * 16;
  for (unsigned e = beg; e < end; ++e) {
    int ec = col[e];
    int n = ec / KNN;
    int k = ec - n * KNN;
    float w = dwb[((size_t)n * KNN + k) * NH];
    float4 g = ld4(tp + ((size_t)n * NH + h) * NC);
    acc = fma4(w, g, acc);
  }
  float* op = out + (size_t)id * 4;
  st4(op, fma4(-c2, acc, ld4(op)));
}

// ------------------------------------------- rhs non-scatter:
//   rhs[t] = Hty[t] + (t>0)*0.5*(rho*phi+gamma)[t]
__global__ void k_rhs(float* __restrict__ rhs, const float* __restrict__ phi,
                      const float* __restrict__ gamma, const float* __restrict__ y,
                      const float* __restrict__ rho, const int* __restrict__ maskp,
                      int it) {
  int id = blockIdx.x * blockDim.x + threadIdx.x;
  if (id >= TNH) return;
  int tn = id >> 2;
  int t = tn / NND;
  float r = rho[it];
  int mask = *maskp;
  float4 acc = zero4();
  if (t > 0) {
    float4 ph = ld4(phi + (size_t)id * 4);
    float4 ga = ld4(gamma + (size_t)id * 4);
    acc = scale4(0.5f, fma4(r, ph, ga));
  }
  if (t < mask) acc = add4(acc, ld4(y + (size_t)tn * NC));
  st4(rhs + (size_t)id * 4, acc);
}

// ------------------------------------------- rhs scatter:
//   rhs[t,m,h] -= 0.5 * sum_{e->m} d_ew * (rho*phi+gamma)[t+1,n_e,h]
__global__ void k_rhs_scat(float* __restrict__ rhs, const float* __restrict__ phi,
                           const float* __restrict__ gamma, const float* __restrict__ d_ew,
                           const unsigned* __restrict__ rowptr, const int* __restrict__ col,
                           const float* __restrict__ rho, int it) {
  int id = blockIdx.x * blockDim.x + threadIdx.x;
  if (id >= SCAT_THR) return;
  int h = id & 3;
  int tm = id >> 2;
  int m = tm % NND;
  int t = tm / NND;
  float r = rho[it];
  float4 acc = zero4();
  unsigned beg = rowptr[m], end = rowptr[m + 1];
  const float* dwb = d_ew + (size_t)t * NND * KNN * NH + h;
  const float* pp = phi + (size_t)(t + 1) * NND * 16;
  const float* gp = gamma + (size_t)(t + 1) * NND * 16;
  for (unsigned e = beg; e < end; ++e) {
    int ec = col[e];
    int n = ec / KNN;
    int k = ec - n * KNN;
    float w = dwb[((size_t)n * KNN + k) * NH];
    size_t off = ((size_t)n * NH + h) * NC;
    float4 s = fma4(r, ld4(pp + off), ld4(gp + off));
    acc = fma4(w, s, acc);
  }
  float* op = rhs + (size_t)id * 4;
  st4(op, fma4(-0.5f, acc, ld4(op)));
}

// ------------------------------------------- CG residual init: r = rhs - Ap; p = r
__global__ void k_r_init(float* __restrict__ r, float* __restrict__ p,
                         const float* __restrict__ rhs, const float* __restrict__ Ap) {
  int id = blockIdx.x * blockDim.x + threadIdx.x;
  if (id >= TNH) return;
  float4 v = sub4(ld4(rhs + (size_t)id * 4), ld4(Ap + (size_t)id * 4));
  st4(r + (size_t)id * 4, v);
  st4(p + (size_t)id * 4, v);
}

// ------------------------------------------- CG update (per-head alpha/beta)
__global__ void k_cg_update(float* __restrict__ x, float* __restrict__ r,
                            float* __restrict__ p, const float* __restrict__ Ap,
                            const float* __restrict__ alpha, const float* __restrict__ beta,
                            int it, int jt) {
  int id = blockIdx.x * blockDim.x + threadIdx.x;
  if (id >= TNH) return;
  int h = id & 3;
  float a = alpha[(it * N_CG + jt) * NH + h];
  float b = beta[(it * N_CG + jt) * NH + h];
  float4 pv = ld4(p + (size_t)id * 4);
  float4 av = ld4(Ap + (size_t)id * 4);
  float4 xv = fma4(a, pv, ld4(x + (size_t)id * 4));
  float4 rv = fma4(-a, av, ld4(r + (size_t)id * 4));
  float4 pn = fma4(b, pv, rv);
  st4(x + (size_t)id * 4, xv);
  st4(r + (size_t)id * 4, rv);
  st4(p + (size_t)id * 4, pn);
}

// ------------------------------------------- soft-threshold + dual update
__global__ void k_shrink(float* __restrict__ phi, float* __restrict__ gamma,
                         const float* __restrict__ ldrx, const float* __restrict__ rho,
                         const float* __restrict__ mu_d1, int it) {
  int id = blockIdx.x * blockDim.x + threadIdx.x;
  if (id >= TNH) return;
  float r = rho[it];
  float thr = mu_d1[it] / r;
  float inv = 1.0f / r;
  float4 lx = ld4(ldrx + (size_t)id * 4);
  float4 ga = ld4(gamma + (size_t)id * 4);
  float4 ph, go;
  {
    float s, u, pv;
    s = lx.x - ga.x * inv; u = fabsf(s) - thr; pv = copysignf(fmaxf(u, 0.f), s);
    ph.x = pv; go.x = ga.x + r * (pv - lx.x);
    s = lx.y - ga.y * inv; u = fabsf(s) - thr; pv = copysignf(fmaxf(u, 0.f), s);
    ph.y = pv; go.y = ga.y + r * (pv - lx.y);
    s = lx.z - ga.z * inv; u = fabsf(s) - thr; pv = copysignf(fmaxf(u, 0.f), s);
    ph.z = pv; go.z = ga.z + r * (pv - lx.z);
    s = lx.w - ga.w * inv; u = fabsf(s) - thr; pv = copysignf(fmaxf(u, 0.f), s);
    ph.w = pv; go.w = ga.w + r * (pv - lx.w);
  }
  st4(phi + (size_t)id * 4, ph);
  st4(gamma + (size_t)id * 4, go);
}

// ------------------------------------------- head combine via V_WMMA_F32_16X16X4_F32
// out[(t,n),c] = sum_h w[h] * x[(t,n),h,c].  M rows = 16 flattened (node,channel)
// pairs (4 nodes), K = 4 heads, B = weights broadcast across 16 columns.
__global__ void k_combine(const float* __restrict__ x, const float* __restrict__ w,
                          float* __restrict__ out) {
  const int G = (T_LEN * NND) / 4;  // WMMA tiles of 4 (t,n) groups
  int gtid = blockIdx.x * blockDim.x + threadIdx.x;
  int wave = gtid >> 5;
  int lane = gtid & 31;
  int nwaves = (gridDim.x * blockDim.x) >> 5;
  float w0 = w[0], w1 = w[1], w2 = w[2], w3 = w[3];
  // B (4x16) layout mirrors 16x4 A: VGPR0 = K0 (lanes 0-15) / K2 (lanes 16-31)
  v2f b;
  b.x = (lane < 16) ? w0 : w2;
  b.y = (lane < 16) ? w1 : w3;
  int mrow = lane & 15;          // A row index M for this lane
  int node_off = mrow >> 2;      // which of the 4 (t,n) groups
  int c = mrow & 3;              // channel
  int hbase = (lane < 16) ? 0 : 2;
  for (int grp = wave; grp < G; grp += nwaves) {
    size_t tn = (size_t)grp * 4 + node_off;
    size_t basea = tn * 16 + (size_t)hbase * 4 + c;  // x[(tn*NH+h)*NC+c]
    v2f a;
    a.x = x[basea];        // head hbase
    a.y = x[basea + 4];    // head hbase+1
    v8f acc = {};
    acc = __builtin_amdgcn_wmma_f32_16x16x4_f32(
        /*neg_a=*/false, a, /*neg_b=*/false, b,
        /*c_mod=*/(short)0, acc, /*reuse_a=*/false, /*reuse_b=*/false);
    // All D columns are identical; lane 0 holds rows 0..7, lane 16 rows 8..15.
    size_t ob = (size_t)grp * 16;
    if (lane == 0) {
#pragma unroll
      for (int j = 0; j < 8; ++j) out[ob + j] = acc[j];
    } else if (lane == 16) {
#pragma unroll
      for (int j = 0; j < 8; ++j) out[ob + 8 + j] = acc[j];
    }
  }
}

// ================================================================ launcher
static inline int blks(int n) { return (n + 255) / 256; }

extern "C" void kernel_launch(void* const* d_in, const int* in_sizes, int n_in,
                              void* d_out, int out_size, void* d_ws, size_t ws_size,
                              hipStream_t stream) {
  (void)in_sizes; (void)n_in; (void)out_size; (void)ws_size;
  const float* y     = (const float*)d_in[0];
  const float* u_ew  = (const float*)d_in[1];
  const float* d_ew  = (const float*)d_in[2];
  const float* mu_u  = (const float*)d_in[3];
  const float* mu_d1 = (const float*)d_in[4];
  const float* mu_d2 = (const float*)d_in[5];
  const float* rho   = (const float*)d_in[6];
  const float* alpha = (const float*)d_in[7];
  const float* beta  = (const float*)d_in[8];
  const float* cw    = (const float*)d_in[9];
  const int*   nn    = (const int*)d_in[10];
  const int*   maskp = (const int*)d_in[11];
  float* out = (float*)d_out;

  // Workspace: 8 state tensors (15.36 MB each, 256B-aligned) + CSR arrays.
  // Total ~= 124 MB; everything stays L2-resident on MI455X (192 MB L2).
  char* w = (char*)d_ws;
  const size_t SB = (size_t)S_ELEMS * sizeof(float);  // multiple of 256
  float* X    = (float*)w; w += SB;
  float* PHI  = (float*)w; w += SB;
  float* GAM  = (float*)w; w += SB;
  float* R    = (float*)w; w += SB;
  float* P    = (float*)w; w += SB;
  float* AP   = (float*)w; w += SB;
  float* TMP  = (float*)w; w += SB;
  float* RHS  = (float*)w; w += SB;
  unsigned* deg    = (unsigned*)w; w += ((NND * 4 + 255) & ~255u);
  unsigned* rowptr = (unsigned*)w; w += (((NND + 1) * 4 + 255) & ~255u);
  unsigned* cursor = (unsigned*)w; w += ((NND * 4 + 255) & ~255u);
  int* col         = (int*)w;      w += ((E_EDGES * 4 + 255) & ~255u);

  // ---- build reverse-CSR adjacency (converts all scatters into owned gathers)
  hipMemsetAsync(deg, 0, (size_t)NND * 4, stream);
  k_count<<<blks(E_EDGES), 256, 0, stream>>>(deg, nn);
  k_scan<<<1, 1024, 0, stream>>>(rowptr, deg);
  hipMemcpyAsync(cursor, rowptr, (size_t)NND * 4, hipMemcpyDeviceToDevice, stream);
  k_fill<<<blks(E_EDGES), 256, 0, stream>>>(col, cursor, nn);

  // ---- ADMM
  k_init<<<blks(TNH), 256, 0, stream>>>(X, GAM, y);
  k_ldr<<<blks(TNH), 256, 0, stream>>>(PHI, X, d_ew, nn);

  for (int i = 0; i < N_ADMM; ++i) {
    // rhs = ldr_t(rho*phi + gamma)/2 + Hty
    k_rhs<<<blks(TNH), 256, 0, stream>>>(RHS, PHI, GAM, y, rho, maskp, i);
    k_rhs_scat<<<blks(SCAT_THR), 256, 0, stream>>>(RHS, PHI, GAM, d_ew, rowptr, col, rho, i);
    // Ap = lhs(x);  r = rhs - Ap;  p = r
    k_ldr<<<blks(TNH), 256, 0, stream>>>(TMP, X, d_ew, nn);
    k_lhs_ns<<<blks(TNH), 256, 0, stream>>>(AP, X, TMP, u_ew, nn, mu_u, mu_d2, rho, maskp, i);
    k_lhs_scat<<<blks(SCAT_THR), 256, 0, stream>>>(AP, TMP, d_ew, rowptr, col, mu_d2, rho, i);
    k_r_init<<<blks(TNH), 256, 0, stream>>>(R, P, RHS, AP);
    for (int j = 0; j < N_CG; ++j) {
      k_ldr<<<blks(TNH), 256, 0, stream>>>(TMP, P, d_ew, nn);
      k_lhs_ns<<<blks(TNH), 256, 0, stream>>>(AP, P, TMP, u_ew, nn, mu_u, mu_d2, rho, maskp, i);
      k_lhs_scat<<<blks(SCAT_THR), 256, 0, stream>>>(AP, TMP, d_ew, rowptr, col, mu_d2, rho, i);
      k_cg_update<<<blks(TNH), 256, 0, stream>>>(X, R, P, AP, alpha, beta, i, j);
    }
    // shrinkage + dual update
    k_ldr<<<blks(TNH), 256, 0, stream>>>(TMP, X, d_ew, nn);
    k_shrink<<<blks(TNH), 256, 0, stream>>>(PHI, GAM, TMP, rho, mu_d1, i);
  }

  // final head combine on the tensor core (f32 WMMA, K = N_HEADS = 4)
  k_combine<<<256, 256, 0, stream>>>(X, cw, out);
}